// ParallelRetention_reduce_25374666785437
// MI455X (gfx1250) — compile-verified
//
#include <hip/hip_runtime.h>
#include <hip/hip_bf16.h>
#include <math.h>

// ParallelRetention forward for MI455X (gfx1250, wave32).
// B=2, H=8, N=2048, DK=DV=64.
//
// Bandwidth-bound on the 268MB omask row-sum (~11.5us @ 23.3TB/s); matmul work
// is only ~8.6 GFLOP, so full fp32 precision via V_WMMA_F32_16X16X4_F32 is free.
// This revision makes every WMMA operand gather a single aligned ds_load_b64
// (contiguous v2f in LDS) to eliminate the v_dual_mov repacking seen in the
// previous round's assembly.

typedef __attribute__((ext_vector_type(2))) float v2f;
typedef __attribute__((ext_vector_type(8))) float v8f;

#define BH   16      // B*H
#define NSEQ 2048
#define DK   64
#define DV   64

// ---------------------------------------------------------------------------
// Kernel 1: per-row omask reduction -> coefficient arrays
//   qcoef[row] = omask[row,0] / sqrt(sum_j omask[row,j])
//   kcoef[row] = 1 / omask[row,0]
// ---------------------------------------------------------------------------
__global__ void __launch_bounds__(256)
pret_coef_kernel(const float* __restrict__ omask,
                 float* __restrict__ qcoef,
                 float* __restrict__ kcoef) {
  __shared__ float red[256];
  const int row = blockIdx.x;                 // 0 .. BH*NSEQ-1
  const size_t base = (size_t)row * NSEQ;
  float s = 0.f;
  #pragma unroll
  for (int j = 0; j < NSEQ / 256; ++j)
    s += omask[base + threadIdx.x + 256 * j];
  red[threadIdx.x] = s;
  __syncthreads();
  for (int off = 128; off > 0; off >>= 1) {
    if (threadIdx.x < off) red[threadIdx.x] += red[threadIdx.x + off];
    __syncthreads();
  }
  if (threadIdx.x == 0) {
    const float o0 = omask[base];
    qcoef[row] = o0 * rsqrtf(red[0]);
    kcoef[row] = 1.0f / o0;
  }
}

// ---------------------------------------------------------------------------
// Kernel 2: blocked causal linear attention with fp32 WMMA.
// One block = one (b,h) x 128 query rows; 8 waves, one 16-row tile per wave.
// ---------------------------------------------------------------------------
__global__ void __launch_bounds__(256)
pret_attn_kernel(const float* __restrict__ q,
                 const float* __restrict__ k,
                 const float* __restrict__ v,
                 const float* __restrict__ qcoef,
                 const float* __restrict__ kcoef,
                 float* __restrict__ out) {
  // Pitches chosen for 8B alignment (even) and conflict-free column reads.
  __shared__ float sK [16][68];     // k_bar key tile, row-major [key][dim]
  __shared__ float sVT[64][18];     // v key tile TRANSPOSED   [dv ][key]
  __shared__ float sS [8][16][18];  // per-wave S spill for C->A layout change

  const int nblkq = NSEQ / 128;     // 16 query blocks per head
  const int bh    = blockIdx.x / nblkq;
  const int qblk  = blockIdx.x % nblkq;
  const int wave  = threadIdx.x >> 5;
  const int lane  = threadIdx.x & 31;
  const int n16   = lane & 15;
  const int half  = lane >> 4;

  const int tile = qblk * 8 + wave; // this wave's query tile index
  const int q0   = tile * 16;       // first query row of the tile

  // Preload A fragments of q_bar (16x64): 16 chunks of K=4.
  // A layout (16x4 f32): lane m=lane%16; VGPR0 = K {0|2}, VGPR1 = K {1|3}.
  v2f a[16];
  {
    const float qcv   = qcoef[bh * NSEQ + q0 + n16];
    const float* qrow = q + ((size_t)(bh * NSEQ + q0 + n16)) * DK;
    #pragma unroll
    for (int c = 0; c < 16; ++c) {
      v2f qv = *(const v2f*)(qrow + 4 * c + 2 * half);  // contiguous pair
      a[c].x = qv.x * qcv;
      a[c].y = qv.y * qcv;
    }
  }

  const v8f zero8 = {0.f, 0.f, 0.f, 0.f, 0.f, 0.f, 0.f, 0.f};
  v8f Oacc[4];
  #pragma unroll
  for (int t = 0; t < 4; ++t) Oacc[t] = zero8;
  v8f Pacc = zero8;                 // rowsum accumulator (all 16 cols equal)

  const v2f ones = {1.0f, 1.0f};    // all-ones B matrix (layout invariant)

  const int ktEnd = qblk * 8 + 8;   // key tiles 0 .. last tile of this block
  for (int kt = 0; kt < ktEnd; ++kt) {
    // ---- cooperative stage: scaled K tile (row-major) + V tile (transposed)
    {
      const int r    = threadIdx.x >> 4;       // key row 0..15
      const int c4   = (threadIdx.x & 15) * 4; // dim col 0..60
      const int krow = kt * 16 + r;
      const float kcv = kcoef[bh * NSEQ + krow];
      const float4 kv = *(const float4*)(k + ((size_t)(bh * NSEQ + krow)) * DK + c4);
      sK[r][c4 + 0] = kv.x * kcv;
      sK[r][c4 + 1] = kv.y * kcv;
      sK[r][c4 + 2] = kv.z * kcv;
      sK[r][c4 + 3] = kv.w * kcv;
      const float4 vv = *(const float4*)(v + ((size_t)(bh * NSEQ + krow)) * DV + c4);
      sVT[c4 + 0][r] = vv.x;        // transpose on store: [dv][key]
      sVT[c4 + 1][r] = vv.y;
      sVT[c4 + 2][r] = vv.z;
      sVT[c4 + 3][r] = vv.w;
    }
    __syncthreads();

    if (kt <= tile) {  // wave-uniform -> EXEC all ones inside (WMMA-legal)
      // ---- S = q_bar(16x64) @ k_bar^T(64x16), fp32 WMMA, 16 K-chunks ----
      // B-frag (k, k+1) of column n is contiguous in sK[n][...] -> ds_load_b64
      v8f S = zero8;
      #pragma unroll
      for (int c = 0; c < 16; ++c) {
        const v2f b = *(const v2f*)&sK[n16][4 * c + 2 * half];
        S = __builtin_amdgcn_wmma_f32_16x16x4_f32(false, a[c], false, b,
                                                  (short)0, S, false, false);
      }

      // ---- causal mask on the diagonal tile (keep j <= i) ----
      if (kt == tile) {
        const int key = kt * 16 + n16;
        #pragma unroll
        for (int g = 0; g < 8; ++g) {
          const int qr = q0 + g + 8 * half;  // C layout: m = g + 8*(lane/16)
          if (key > qr) S[g] = 0.0f;
        }
      }

      // ---- spill S (C layout) to LDS, reload as A fragments (4 K-chunks) ----
      #pragma unroll
      for (int g = 0; g < 8; ++g) sS[wave][g + 8 * half][n16] = S[g];
      v2f sa[4];
      #pragma unroll
      for (int c = 0; c < 4; ++c)
        sa[c] = *(const v2f*)&sS[wave][n16][4 * c + 2 * half];  // ds_load_b64

      // ---- P += rowsum(S) via all-ones B matrix ----
      #pragma unroll
      for (int c = 0; c < 4; ++c)
        Pacc = __builtin_amdgcn_wmma_f32_16x16x4_f32(false, sa[c], false, ones,
                                                     (short)0, Pacc, false, false);

      // ---- O += S(16x16) @ V(16x64) : 4 column tiles x 4 K-chunks ----
      // V is transposed in LDS, so (k, k+1) of column n is contiguous.
      #pragma unroll
      for (int ct = 0; ct < 4; ++ct) {
        #pragma unroll
        for (int c = 0; c < 4; ++c) {
          const v2f vb = *(const v2f*)&sVT[ct * 16 + n16][4 * c + 2 * half];
          Oacc[ct] = __builtin_amdgcn_wmma_f32_16x16x4_f32(false, sa[c], false, vb,
                                                           (short)0, Oacc[ct], false, false);
        }
      }
    }
    __syncthreads();
  }

  // ---- epilogue: O / max(|P|, 1).  P's C-layout rows align elementwise with O. ----
  #pragma unroll
  for (int ct = 0; ct < 4; ++ct) {
    #pragma unroll
    for (int g = 0; g < 8; ++g) {
      const float p = fmaxf(fabsf(Pacc[g]), 1.0f);
      out[((size_t)(bh * NSEQ + q0 + g + 8 * half)) * DV + ct * 16 + n16] =
          Oacc[ct][g] / p;
    }
  }
}

// ---------------------------------------------------------------------------
extern "C" void kernel_launch(void* const* d_in, const int* in_sizes, int n_in,
                              void* d_out, int out_size, void* d_ws, size_t ws_size,
                              hipStream_t stream) {
  const float* q     = (const float*)d_in[0];
  const float* k     = (const float*)d_in[1];
  const float* v     = (const float*)d_in[2];
  const float* omask = (const float*)d_in[3];
  float* out = (float*)d_out;

  float* qcoef = (float*)d_ws;                  // BH*NSEQ floats
  float* kcoef = qcoef + (size_t)BH * NSEQ;     // BH*NSEQ floats

  pret_coef_kernel<<<BH * NSEQ, 256, 0, stream>>>(omask, qcoef, kcoef);
  pret_attn_kernel<<<BH * (NSEQ / 128), 256, 0, stream>>>(q, k, v,
                                                          qcoef, kcoef, out);
}